// SHM_30167850287955
// MI455X (gfx1250) — compile-verified
//
#include <hip/hip_runtime.h>
#include <hip/hip_bf16.h>

typedef float v2f __attribute__((ext_vector_type(2)));
typedef float v8f __attribute__((ext_vector_type(8)));

#define Bb 8
#define Tt 1024
#define Dd 512
#define Mm 64
#define Oo 512

// ---------------------------------------------------------------------------
// Kernel 1: LayerNorm over D=512 per row + eta = sigmoid(xn . We)
// One wave per row (16 elements/lane), 8 waves per block, no LDS, shuffle reductions.
// ---------------------------------------------------------------------------
__global__ void ln_eta_kernel(const float* __restrict__ x,
                              const float* __restrict__ gamma,
                              const float* __restrict__ beta,
                              const float* __restrict__ We,
                              float* __restrict__ xn,
                              float* __restrict__ eta)
{
    const int wave = threadIdx.x >> 5;
    const int lane = threadIdx.x & 31;
    const int row  = blockIdx.x * 8 + wave;          // 0..8191
    const float* xr = x + (size_t)row * Dd;

    float v[16];
    float sum = 0.f, sq = 0.f;
#pragma unroll
    for (int e = 0; e < 16; ++e) {
        float t = xr[lane + 32 * e];
        v[e] = t; sum += t; sq += t * t;
    }
#pragma unroll
    for (int m = 16; m >= 1; m >>= 1) {
        sum += __shfl_xor(sum, m, 32);
        sq  += __shfl_xor(sq,  m, 32);
    }
    const float mu = sum * (1.0f / Dd);
    const float var = sq * (1.0f / Dd) - mu * mu;
    const float rs = rsqrtf(var + 1e-5f);

    float dot = 0.f;
    float* xo = xn + (size_t)row * Dd;
#pragma unroll
    for (int e = 0; e < 16; ++e) {
        const int d = lane + 32 * e;
        float xv = (v[e] - mu) * rs * gamma[d] + beta[d];
        xo[d] = xv;
        dot += xv * We[d];
    }
#pragma unroll
    for (int m = 16; m >= 1; m >>= 1) dot += __shfl_xor(dot, m, 32);
    if (lane == 0) eta[row] = 1.0f / (1.0f + expf(-dot));
}

// ---------------------------------------------------------------------------
// Kernel 2: projections  Out[8192,64] = xn[8192,512] @ W[64,512]^T
// One wave (32 threads) per block computes a 16x64 strip via fp32 WMMA 16x16x4,
// chaining 128 WMMAs over K=512. For K/Q (weight 0/1): fused relu + L1-norm.
// ---------------------------------------------------------------------------
__global__ void proj_gemm_kernel(const float* __restrict__ xn,
                                 const float* __restrict__ Wk,
                                 const float* __restrict__ Wq,
                                 const float* __restrict__ Wv,
                                 const float* __restrict__ Wvc,
                                 float* __restrict__ Ko,
                                 float* __restrict__ Qo,
                                 float* __restrict__ Vo,
                                 float* __restrict__ vco)
{
    const int w = blockIdx.y;                         // 0=K 1=Q 2=V 3=vc
    const float* W = (w == 0) ? Wk : (w == 1) ? Wq : (w == 2) ? Wv : Wvc;
    float* Out     = (w == 0) ? Ko : (w == 1) ? Qo : (w == 2) ? Vo : vco;

    const int tileM = blockIdx.x * 16;
    const int lane  = threadIdx.x & 31;
    const int hi    = lane >> 4;                      // 0: K=0,1  1: K=2,3
    const int lr    = lane & 15;

    const float* Arow = xn + (size_t)(tileM + lr) * Dd + 2 * hi;
    const float* B0 = W + (size_t)(lr +  0) * Dd + 2 * hi;
    const float* B1 = W + (size_t)(lr + 16) * Dd + 2 * hi;
    const float* B2 = W + (size_t)(lr + 32) * Dd + 2 * hi;
    const float* B3 = W + (size_t)(lr + 48) * Dd + 2 * hi;

    v8f c0 = {}, c1 = {}, c2 = {}, c3 = {};
    for (int kk = 0; kk < Dd; kk += 4) {
        v2f a  = *(const v2f*)(Arow + kk);
        v2f b0 = *(const v2f*)(B0 + kk);
        v2f b1 = *(const v2f*)(B1 + kk);
        v2f b2 = *(const v2f*)(B2 + kk);
        v2f b3 = *(const v2f*)(B3 + kk);
        c0 = __builtin_amdgcn_wmma_f32_16x16x4_f32(false, a, false, b0, (short)0, c0, false, false);
        c1 = __builtin_amdgcn_wmma_f32_16x16x4_f32(false, a, false, b1, (short)0, c1, false, false);
        c2 = __builtin_amdgcn_wmma_f32_16x16x4_f32(false, a, false, b2, (short)0, c2, false, false);
        c3 = __builtin_amdgcn_wmma_f32_16x16x4_f32(false, a, false, b3, (short)0, c3, false, false);
    }

    if (w < 2) {
        // relu + L1 normalize each row of 64 (row r+8*hi lives in c*[r] of this half-wave)
#pragma unroll
        for (int r = 0; r < 8; ++r) {
            c0[r] = fmaxf(c0[r], 0.f); c1[r] = fmaxf(c1[r], 0.f);
            c2[r] = fmaxf(c2[r], 0.f); c3[r] = fmaxf(c3[r], 0.f);
            float s = c0[r] + c1[r] + c2[r] + c3[r];
#pragma unroll
            for (int m = 8; m >= 1; m >>= 1) s += __shfl_xor(s, m, 32);  // 16-lane halves
            float inv = 1.0f / (1e-5f + s);
            c0[r] *= inv; c1[r] *= inv; c2[r] *= inv; c3[r] *= inv;
        }
    }

#pragma unroll
    for (int r = 0; r < 8; ++r) {
        float* o = Out + (size_t)(tileM + r + 8 * hi) * Mm + lr;
        o[0]  = c0[r];
        o[16] = c1[r];
        o[32] = c2[r];
        o[48] = c3[r];
    }
}

// ---------------------------------------------------------------------------
// Kernel 3: sequential scan.  Rows of the state are independent:
//   state[i,j] <- state[i,j]*(1+tanh(theta_i*vc_j)) + eta*V_i*K_j
//   y[i] = sum_j state[i,j]*Q_j
// One wave per (b, i): lane holds state[i,lane] and state[i,lane+32].
// No barriers — waves free-run their private recurrences.
// ---------------------------------------------------------------------------
__global__ void scan_kernel(const float* __restrict__ K,
                            const float* __restrict__ Q,
                            const float* __restrict__ V,
                            const float* __restrict__ vc,
                            const float* __restrict__ eta,
                            const float* __restrict__ theta_matrix,
                            float* __restrict__ y)
{
    const int b    = blockIdx.x;                      // 0..7
    const int wave = threadIdx.x >> 5;                // 0..7
    const int lane = threadIdx.x & 31;
    const int i    = blockIdx.y * 8 + wave;           // 0..63
    const float theta_i = theta_matrix[i];            // row 0 of theta_matrix

    float s0 = 0.f, s1 = 0.f;
    const size_t base = (size_t)b * Tt * Mm;

    for (int t = 0; t < Tt; ++t) {
        const size_t o = base + (size_t)t * Mm;
        const float vc0 = vc[o + lane], vc1 = vc[o + lane + 32];
        const float k0  = K[o + lane],  k1  = K[o + lane + 32];
        const float q0  = Q[o + lane],  q1  = Q[o + lane + 32];
        const float et  = eta[b * Tt + t];
        const float vi  = V[o + i];

        const float cc0 = 1.0f + tanhf(theta_i * vc0);
        const float cc1 = 1.0f + tanhf(theta_i * vc1);
        const float u   = et * vi;
        s0 = s0 * cc0 + u * k0;
        s1 = s1 * cc1 + u * k1;

        float p = s0 * q0 + s1 * q1;
#pragma unroll
        for (int m = 16; m >= 1; m >>= 1) p += __shfl_xor(p, m, 32);
        if (lane == 0) y[o + i] = p;
    }
}

// ---------------------------------------------------------------------------
// Kernel 4: out[8192,512] = y[8192,64] @ Wout[512,64]^T + bout  (fp32 WMMA)
// One wave per 16x64 strip; 16 k-steps.
// ---------------------------------------------------------------------------
__global__ void out_gemm_kernel(const float* __restrict__ y,
                                const float* __restrict__ Wout,
                                const float* __restrict__ bout,
                                float* __restrict__ out)
{
    const int tileM = blockIdx.x * 16;                // 512 blocks
    const int tileN = blockIdx.y * 64;                // 8 blocks
    const int lane  = threadIdx.x & 31;
    const int hi    = lane >> 4;
    const int lr    = lane & 15;

    const float* Arow = y + (size_t)(tileM + lr) * Mm + 2 * hi;
    const float* B0 = Wout + (size_t)(tileN + lr +  0) * Mm + 2 * hi;
    const float* B1 = Wout + (size_t)(tileN + lr + 16) * Mm + 2 * hi;
    const float* B2 = Wout + (size_t)(tileN + lr + 32) * Mm + 2 * hi;
    const float* B3 = Wout + (size_t)(tileN + lr + 48) * Mm + 2 * hi;

    v8f c0 = {}, c1 = {}, c2 = {}, c3 = {};
#pragma unroll
    for (int kk = 0; kk < Mm; kk += 4) {
        v2f a  = *(const v2f*)(Arow + kk);
        v2f b0 = *(const v2f*)(B0 + kk);
        v2f b1 = *(const v2f*)(B1 + kk);
        v2f b2 = *(const v2f*)(B2 + kk);
        v2f b3 = *(const v2f*)(B3 + kk);
        c0 = __builtin_amdgcn_wmma_f32_16x16x4_f32(false, a, false, b0, (short)0, c0, false, false);
        c1 = __builtin_amdgcn_wmma_f32_16x16x4_f32(false, a, false, b1, (short)0, c1, false, false);
        c2 = __builtin_amdgcn_wmma_f32_16x16x4_f32(false, a, false, b2, (short)0, c2, false, false);
        c3 = __builtin_amdgcn_wmma_f32_16x16x4_f32(false, a, false, b3, (short)0, c3, false, false);
    }

    const float bb0 = bout[tileN + lr +  0];
    const float bb1 = bout[tileN + lr + 16];
    const float bb2 = bout[tileN + lr + 32];
    const float bb3 = bout[tileN + lr + 48];

#pragma unroll
    for (int r = 0; r < 8; ++r) {
        float* o = out + (size_t)(tileM + r + 8 * hi) * Oo + tileN + lr;
        o[0]  = c0[r] + bb0;
        o[16] = c1[r] + bb1;
        o[32] = c2[r] + bb2;
        o[48] = c3[r] + bb3;
    }
}

// ---------------------------------------------------------------------------
extern "C" void kernel_launch(void* const* d_in, const int* in_sizes, int n_in,
                              void* d_out, int out_size, void* d_ws, size_t ws_size,
                              hipStream_t stream)
{
    const float* x     = (const float*)d_in[0];
    const float* Wk    = (const float*)d_in[1];
    const float* Wq    = (const float*)d_in[2];
    const float* Wv    = (const float*)d_in[3];
    const float* Wvc   = (const float*)d_in[4];
    const float* We    = (const float*)d_in[5];
    const float* theta = (const float*)d_in[6];
    const float* Wout  = (const float*)d_in[7];
    const float* bout  = (const float*)d_in[8];
    const float* gamma = (const float*)d_in[9];
    const float* beta  = (const float*)d_in[10];
    float* out = (float*)d_out;

    float* ws  = (float*)d_ws;
    const size_t NROW = (size_t)Bb * Tt;              // 8192
    float* xn  = ws;                                  // 8192*512
    float* Ko  = xn  + NROW * Dd;                     // 8192*64
    float* Qo  = Ko  + NROW * Mm;
    float* Vo  = Qo  + NROW * Mm;
    float* vco = Vo  + NROW * Mm;
    float* yb  = vco + NROW * Mm;
    float* eta = yb  + NROW * Mm;                     // 8192

    ln_eta_kernel<<<dim3(NROW / 8), dim3(256), 0, stream>>>(x, gamma, beta, We, xn, eta);
    proj_gemm_kernel<<<dim3(NROW / 16, 4), dim3(32), 0, stream>>>(xn, Wk, Wq, Wv, Wvc,
                                                                  Ko, Qo, Vo, vco);
    scan_kernel<<<dim3(Bb, Mm / 8), dim3(256), 0, stream>>>(Ko, Qo, Vo, vco, eta, theta, yb);
    out_gemm_kernel<<<dim3(NROW / 16, Oo / 64), dim3(32), 0, stream>>>(yb, Wout, bout, out);
}